// MyDenseLayer_541165879877
// MI455X (gfx1250) — compile-verified
//
#include <hip/hip_runtime.h>

typedef __attribute__((ext_vector_type(2))) float v2f;
typedef __attribute__((ext_vector_type(4))) float v4f;
typedef __attribute__((ext_vector_type(8))) float v8f;

// Vector quantization: nearest-of-4 codeword lookup for dim-2 points.
// Memory-bound: 128 MB in + 128 MB out => ~11.5 us floor at 23.3 TB/s.
// Score matmul (x @ vq.T with K=2) done exactly in fp32 with
// V_WMMA_F32_16X16X4_F32:
//   A (16x4): rows 0..3 = codewords (K=0,1 = dims; K=2,3 and rows 4+ zero)
//   B (4x16): column n = a point; lane n holds (x0,x1), lanes 16..31 = 0
//   C bias  : row m = -0.5*||c_m||^2  =>  D[m][n] = c_m.x_n - 0.5||c_m||^2
// C/D layout puts D[0..3] for column n entirely in lane n, so argmax is an
// in-lane compare chain. Column->point mapping is chosen so each lane's b128
// load (2 points) feeds a WMMA *pair* and both results pack into one b128
// store from the same lane: full 256 B per memory instruction, loads batched
// up front for memory-level parallelism.
__global__ __launch_bounds__(256) void vq_nearest_wmma(
    const v4f* __restrict__ x4,    // nPoints/2 entries, 2 points each
    const float* __restrict__ vq,  // 4 x 2 f32 codebook
    v4f* __restrict__ out4,        // nPoints/2 entries
    int nPoints)
{
    const int lane   = threadIdx.x & 31;
    const int lane16 = lane & 15;
    const bool isLo  = lane < 16;

    const int wavesPerBlk = blockDim.x >> 5;
    const int waveId = blockIdx.x * wavesPerBlk + (threadIdx.x >> 5);
    const int nWaves = gridDim.x * wavesPerBlk;

    // Codebook in registers (wave-uniform -> SGPRs).
    const float c00 = vq[0], c01 = vq[1];
    const float c10 = vq[2], c11 = vq[3];
    const float c20 = vq[4], c21 = vq[5];
    const float c30 = vq[6], c31 = vq[7];

    // A: lane m (<4) holds codeword m; everything else zero (incl. lanes
    // 16..31, which are K=2,3 of rows 0..15).
    v2f A = {0.0f, 0.0f};
    if (lane < 4) {
        A.x = vq[2 * lane];
        A.y = vq[2 * lane + 1];
    }

    // C bias: accumulator VGPR m = -0.5*||c_m||^2 (rows 4..7 unused).
    v8f C = {};
    C[0] = -0.5f * (c00 * c00 + c01 * c01);
    C[1] = -0.5f * (c10 * c10 + c11 * c11);
    C[2] = -0.5f * (c20 * c20 + c21 * c21);
    C[3] = -0.5f * (c30 * c30 + c31 * c31);

    const int nTiles = nPoints >> 7;  // 128 points per wave-tile

    for (int tile = waveId; tile < nTiles; tile += nWaves) {
        const int base2 = tile * 64;  // index in v4f units (2 points each)

        // Batch all 4 b128 NT loads first: 4 x 256 B in flight per wave.
        // Lanes 16..31 mirror lanes 0..15 (same cachelines, coalesced).
        v4f l[4];
#pragma unroll
        for (int j = 0; j < 4; ++j)
            l[j] = __builtin_nontemporal_load(x4 + (base2 + 16 * j + lane16));

        v4f s[4];
#pragma unroll
        for (int j = 0; j < 4; ++j) {
            // WMMA pair for load j: B0 = even points, B1 = odd points.
            v2f B0, B1;
            B0.x = isLo ? l[j].x : 0.0f;
            B0.y = isLo ? l[j].y : 0.0f;
            B1.x = isLo ? l[j].z : 0.0f;
            B1.y = isLo ? l[j].w : 0.0f;

            v8f D0 = __builtin_amdgcn_wmma_f32_16x16x4_f32(
                false, A, false, B0, (short)0, C, false, false);
            v8f D1 = __builtin_amdgcn_wmma_f32_16x16x4_f32(
                false, A, false, B1, (short)0, C, false, false);

            // First-max argmax (strict '>' keeps earliest index, matching
            // jnp.argmax on -distances); select codeword directly.
            float b0 = D0[0], qx0 = c00, qy0 = c01;
            if (D0[1] > b0) { b0 = D0[1]; qx0 = c10; qy0 = c11; }
            if (D0[2] > b0) { b0 = D0[2]; qx0 = c20; qy0 = c21; }
            if (D0[3] > b0) { b0 = D0[3]; qx0 = c30; qy0 = c31; }

            float b1 = D1[0], qx1 = c00, qy1 = c01;
            if (D1[1] > b1) { b1 = D1[1]; qx1 = c10; qy1 = c11; }
            if (D1[2] > b1) { b1 = D1[2]; qx1 = c20; qy1 = c21; }
            if (D1[3] > b1) { b1 = D1[3]; qx1 = c30; qy1 = c31; }

            s[j].x = qx0; s[j].y = qy0; s[j].z = qx1; s[j].w = qy1;
        }

        // 4 b128 NT stores from lanes 0..15 only (256 B contiguous each).
        if (isLo) {
#pragma unroll
            for (int j = 0; j < 4; ++j)
                __builtin_nontemporal_store(
                    s[j], out4 + (base2 + 16 * j + lane16));
        }
    }

    // Tail (nPoints % 128): plain VALU path on one wave (WMMA needs full
    // EXEC, so ragged edges use scalar math). Empty for N = 16.8M.
    const int tailStart = nTiles << 7;
    if (blockIdx.x == 0 && threadIdx.x < 32) {
        const v2f* x2 = (const v2f*)x4;
        v2f* out2 = (v2f*)out4;
        for (int p = tailStart + lane; p < nPoints; p += 32) {
            v2f xv = x2[p];
            float s0 = xv.x * c00 + xv.y * c01 + C[0];
            float s1 = xv.x * c10 + xv.y * c11 + C[1];
            float s2 = xv.x * c20 + xv.y * c21 + C[2];
            float s3 = xv.x * c30 + xv.y * c31 + C[3];
            float b = s0, qx = c00, qy = c01;
            if (s1 > b) { b = s1; qx = c10; qy = c11; }
            if (s2 > b) { b = s2; qx = c20; qy = c21; }
            if (s3 > b) { b = s3; qx = c30; qy = c31; }
            v2f q; q.x = qx; q.y = qy;
            out2[p] = q;
        }
    }
}

extern "C" void kernel_launch(void* const* d_in, const int* in_sizes, int n_in,
                              void* d_out, int out_size, void* d_ws, size_t ws_size,
                              hipStream_t stream) {
    const float* x  = (const float*)d_in[0];  // (4096*4096, 2) f32, flat
    const float* vq = (const float*)d_in[1];  // (4, 2) f32
    float* out = (float*)d_out;               // (N, 2) f32, flat

    const int nPoints = in_sizes[0] / 2;

    // 4096 blocks x 8 waves = 32768 waves; 131072 tiles -> 4 tiles/wave.
    dim3 block(256);
    dim3 grid(4096);
    vq_nearest_wmma<<<grid, block, 0, stream>>>(
        (const v4f*)x, vq, (v4f*)out, nPoints);
}